// GraphUNetVerySmallLessLayers_43018392436832
// MI455X (gfx1250) — compile-verified
//
#include <hip/hip_runtime.h>
#include <hip/hip_bf16.h>

// -------------------- problem constants (from reference) --------------------
static constexpr int BB  = 2;
static constexpr int P0  = 200000;   // N0
static constexpr int P1  = 50000;    // N1
static constexpr int P2  = 12500;    // N2
static constexpr int P3  = 3125;     // N3
static constexpr int NFc = 600000;   // faces
static constexpr int NCc = 400000;   // cell centers

typedef __attribute__((ext_vector_type(2))) float v2f;
typedef __attribute__((ext_vector_type(8))) float v8f;

// -------------------- elementwise / reduction kernels --------------------

__global__ void k_zero(float* __restrict__ p, size_t n) {
  size_t i = (size_t)blockIdx.x * blockDim.x + threadIdx.x;
  if (i < n) p[i] = 0.f;
}

// One block per (b,c); reduce mean/var over node axis. stats[2*bc] = mean, [2*bc+1] = rstd.
__global__ void k_inorm_stats(const float* __restrict__ x, float* __restrict__ stats,
                              int N, int C) {
  int bc = blockIdx.x;
  int b = bc / C, c = bc % C;
  const float* xp = x + (size_t)b * N * C + c;
  float s = 0.f, ss = 0.f;
  for (int i = threadIdx.x; i < N; i += blockDim.x) {
    float v = xp[(size_t)i * C];
    s += v; ss += v * v;
  }
  __shared__ float rs[256], rq[256];
  rs[threadIdx.x] = s; rq[threadIdx.x] = ss;
  __syncthreads();
  for (int w = 128; w > 0; w >>= 1) {
    if (threadIdx.x < w) { rs[threadIdx.x] += rs[threadIdx.x + w]; rq[threadIdx.x] += rq[threadIdx.x + w]; }
    __syncthreads();
  }
  if (threadIdx.x == 0) {
    float m   = rs[0] / (float)N;
    float var = rq[0] / (float)N - m * m;
    stats[2 * bc]     = m;
    stats[2 * bc + 1] = rsqrtf(var + 1e-5f);
  }
}

__global__ void k_inorm_apply(const float* __restrict__ x, const float* __restrict__ stats,
                              float* __restrict__ y, int N, int C, size_t total) {
  size_t i = (size_t)blockIdx.x * blockDim.x + threadIdx.x;
  if (i >= total) return;
  int c = (int)(i % C);
  int b = (int)((i / C) / N);
  int bc = b * C + c;
  y[i] = (x[i] - stats[2 * bc]) * stats[2 * bc + 1];
}

// Edge gather * edge_attr -> atomic scatter-add. Thread t = (edge, channel); both batches.
// Neighboring threads hit consecutive addresses of out[dst*C + c] -> coalesced f32 atomics.
__global__ void k_gs(const float* __restrict__ x, const int* __restrict__ src,
                     const int* __restrict__ dst, const float* __restrict__ ea,
                     float* __restrict__ out, int E, int C, int Ns, int Nd) {
  int t = blockIdx.x * blockDim.x + threadIdx.x;
  if (t >= E * C) return;
  int e = t / C, c = t - e * C;
  int s = src[e], d = dst[e];
  float a = ea[e];
#pragma unroll
  for (int b = 0; b < BB; ++b) {
    float v = x[(size_t)b * Ns * C + (size_t)s * C + c] * a;
    atomicAdd(&out[(size_t)b * Nd * C + (size_t)d * C + c], v);
  }
}

// y[b,n,0:Ca] = a ; y[b,n,Ca:Ca+Cb] = bsrc  (feature-dim concat)
__global__ void k_concat2(const float* __restrict__ a, const float* __restrict__ bsrc,
                          float* __restrict__ y, int Ca, int Cb, size_t total) {
  size_t i = (size_t)blockIdx.x * blockDim.x + threadIdx.x;
  if (i >= total) return;
  int C = Ca + Cb;
  int c = (int)(i % C);
  size_t bn = i / C;
  y[i] = (c < Ca) ? a[bn * Ca + c] : bsrc[bn * Cb + (c - Ca)];
}

// -------------------- WMMA linear --------------------
// Fragment loaders: unconditional clamped loads + v_cndmask data masking.
// kr is always even; rows are 8B-aligned (K even, buffers 8B-aligned) -> v2f load is legal.

template <int K>
__device__ inline v2f load_a_frag(const float* __restrict__ xp, int kr, bool mOK) {
  if constexpr (K % 4 == 0) {
    v2f a = *(const v2f*)(xp + kr);                 // global_load_b64, no branch
    a.x = mOK ? a.x : 0.f;
    a.y = mOK ? a.y : 0.f;
    return a;
  } else {                                          // K == 2 case
    int krc = (kr + 1 < K) ? kr : 0;                // clamp in-bounds
    v2f a = *(const v2f*)(xp + krc);
    bool ok = mOK && (kr < K);                      // kr even, K even: both lanes same validity
    a.x = ok ? a.x : 0.f;
    a.y = ok ? a.y : 0.f;
    return a;
  }
}

template <int K>
__device__ inline v2f load_w_frag(const float* __restrict__ W, int kr, int n, int Cout, bool nOK) {
  int nc = nOK ? n : 0;                             // clamp column in-bounds
  if constexpr (K % 4 == 0) {
    v2f w;
    w.x = W[(size_t)kr * Cout + nc];
    w.y = W[(size_t)(kr + 1) * Cout + nc];
    w.x = nOK ? w.x : 0.f;
    w.y = nOK ? w.y : 0.f;
    return w;
  } else {
    int kr0 = (kr < K) ? kr : 0;
    int kr1 = (kr + 1 < K) ? kr + 1 : 0;
    v2f w;
    w.x = W[(size_t)kr0 * Cout + nc];
    w.y = W[(size_t)kr1 * Cout + nc];
    w.x = (nOK && kr < K) ? w.x : 0.f;
    w.y = (nOK && kr + 1 < K) ? w.y : 0.f;
    return w;
  }
}

// y = act( x1 @ W1 [+ x2 @ W2] + bias ), x*: [B,N,K*] row-major, W*: [K*,Cout] row-major.
// One wave32 per (batch, 16-node tile, 16-outch tile); K fully unrolled -> straight WMMA chain.
template <int K1, int K2>
__global__ __launch_bounds__(32)
void k_linear_wmma(const float* __restrict__ x1, const float* __restrict__ W1,
                   const float* __restrict__ x2, const float* __restrict__ W2,
                   const float* __restrict__ bias, float* __restrict__ y,
                   int N, int Cout, int tilesPerBatch, int doRelu) {
  int lane = threadIdx.x & 31;
  int half = lane >> 4;        // 0: lanes 0-15, 1: lanes 16-31
  int lm   = lane & 15;
  int b       = blockIdx.x / tilesPerBatch;
  int tile    = blockIdx.x - b * tilesPerBatch;
  int rowBase = tile * 16;
  int colBase = blockIdx.y * 16;
  int m = rowBase + lm;                 // A-fragment row for this lane
  int n = colBase + lm;                 // B/D-fragment column for this lane
  bool mOK = (m < N);
  bool nOK = (n < Cout);
  int mClamped = mOK ? m : 0;

  v8f acc = {0.f, 0.f, 0.f, 0.f, 0.f, 0.f, 0.f, 0.f};

  {
    const float* xp = x1 + (size_t)b * N * K1 + (size_t)mClamped * K1;
#pragma unroll
    for (int k0 = 0; k0 < K1; k0 += 4) {
      int kr = k0 + 2 * half;
      v2f a = load_a_frag<K1>(xp, kr, mOK);
      v2f w = load_w_frag<K1>(W1, kr, n, Cout, nOK);
      acc = __builtin_amdgcn_wmma_f32_16x16x4_f32(false, a, false, w, (short)0, acc, false, false);
    }
  }
  if constexpr (K2 > 0) {
    const float* xp = x2 + (size_t)b * N * K2 + (size_t)mClamped * K2;
#pragma unroll
    for (int k0 = 0; k0 < K2; k0 += 4) {
      int kr = k0 + 2 * half;
      v2f a = load_a_frag<K2>(xp, kr, mOK);
      v2f w = load_w_frag<K2>(W2, kr, n, Cout, nOK);
      acc = __builtin_amdgcn_wmma_f32_16x16x4_f32(false, a, false, w, (short)0, acc, false, false);
    }
  }

  float bv = nOK ? bias[nOK ? n : 0] : 0.f;
#pragma unroll
  for (int r = 0; r < 8; ++r) {
    int row = rowBase + r + 8 * half;   // D layout: vgpr r, lane half -> M = r + 8*half
    if (row < N && nOK) {
      float v = acc[r] + bv;
      if (doRelu) v = fmaxf(v, 0.f);
      y[(size_t)b * N * Cout + (size_t)row * Cout + n] = v;
    }
  }
}

// -------------------- host orchestration --------------------

static inline int cdiv(int a, int b) { return (a + b - 1) / b; }

// edge-type order = espec insertion order in setup_inputs()
enum { E_CF = 0, E_FP, E_PC, E_PP0, E_PP1, E_PP2, E_PP3,
       E_POOL0, E_UNPOOL0, E_POOL1, E_UNPOOL1, E_POOL2, E_UNPOOL2, NEDGE };

extern "C" void kernel_launch(void* const* d_in, const int* in_sizes, int n_in,
                              void* d_out, int out_size, void* d_ws, size_t ws_size,
                              hipStream_t stream) {
  // ---- input unpacking (flat dict order) ----
  const float* xcc = (const float*)d_in[0];   // [B,NC,2]
  const float* xf  = (const float*)d_in[1];   // [B,NF,4]
  const int*   esrc[NEDGE]; const int* edst[NEDGE];
  const float* eatt[NEDGE]; int ecnt[NEDGE];
  for (int i = 0; i < NEDGE; ++i) {
    esrc[i] = (const int*)d_in[2 + 2 * i];
    edst[i] = (const int*)d_in[3 + 2 * i];
    eatt[i] = (const float*)d_in[28 + i];
    ecnt[i] = in_sizes[28 + i];
  }
  auto P = [&](int i) { return (const float*)d_in[i]; };
  // params: 41/42 c1_cf W,b | 43/44 c1_fp | 45-47 c1_pp | 48-50 c2_l1 | 51-53 c2_l2
  // 54-56 c3_l1 | 57-59 c3_l2 | 60-62 c4_l1 | 63-65 c4_l2 | 66-68 c4_l3 | 69-71 c4_l4
  // 72-74 c7_l1 | 75-77 c7_l2 | 78-80 c8_l1 | 81-83 c8_l2 | 84-86 c9_l1
  // 87/88 c9_l2 W,b | 89/90 final W,b

  // ---- workspace bump arena ----
  size_t off = 0;
  auto alloc = [&](size_t elems) { float* p = (float*)d_ws + off; off += elems; return p; };
  float* stats = alloc(256);
  // persistent skip tensors
  float* c1 = alloc((size_t)BB * P0 * 12);
  float* c2 = alloc((size_t)BB * P1 * 12);
  float* c3 = alloc((size_t)BB * P2 * 12);
  float* c4 = alloc((size_t)BB * P2 * 24);
  float* c7 = alloc((size_t)BB * P1 * 24);
  float* c8 = alloc((size_t)BB * P0 * 12);
  const size_t arena = off;

  // ---- helpers ----
  auto zero = [&](float* p, size_t n) {
    k_zero<<<(int)((n + 255) / 256), 256, 0, stream>>>(p, n);
  };
  auto inorm = [&](const float* x, float* y, int N, int C) {
    k_inorm_stats<<<BB * C, 256, 0, stream>>>(x, stats, N, C);
    size_t tot = (size_t)BB * N * C;
    k_inorm_apply<<<(int)((tot + 255) / 256), 256, 0, stream>>>(x, stats, y, N, C, tot);
  };
  auto gs = [&](const float* x, int ei, float* out, int C, int Ns, int Nd) {
    zero(out, (size_t)BB * Nd * C);
    int tot = ecnt[ei] * C;
    k_gs<<<cdiv(tot, 256), 256, 0, stream>>>(x, esrc[ei], edst[ei], eatt[ei], out,
                                             ecnt[ei], C, Ns, Nd);
  };
  auto lin = [&](const float* x1, const float* W1, int K1,
                 const float* x2, const float* W2, int K2,
                 const float* bias, float* y, int N, int Cout, int relu) {
    int tpb = cdiv(N, 16);
    dim3 g(BB * tpb, cdiv(Cout, 16));
    if (K1 == 2 && K2 == 0)
      k_linear_wmma<2, 0><<<g, 32, 0, stream>>>(x1, W1, x2, W2, bias, y, N, Cout, tpb, relu);
    else if (K1 == 12 && K2 == 0)
      k_linear_wmma<12, 0><<<g, 32, 0, stream>>>(x1, W1, x2, W2, bias, y, N, Cout, tpb, relu);
    else if (K1 == 12 && K2 == 12)
      k_linear_wmma<12, 12><<<g, 32, 0, stream>>>(x1, W1, x2, W2, bias, y, N, Cout, tpb, relu);
    else if (K1 == 24 && K2 == 24)
      k_linear_wmma<24, 24><<<g, 32, 0, stream>>>(x1, W1, x2, W2, bias, y, N, Cout, tpb, relu);
    else if (K1 == 36 && K2 == 36)
      k_linear_wmma<36, 36><<<g, 32, 0, stream>>>(x1, W1, x2, W2, bias, y, N, Cout, tpb, relu);
  };
  auto cat = [&](const float* a, const float* b, float* y, int N, int Ca, int Cb) {
    size_t tot = (size_t)BB * N * (Ca + Cb);
    k_concat2<<<(int)((tot + 255) / 256), 256, 0, stream>>>(a, b, y, Ca, Cb, tot);
  };
  // _pp: y = relu(x@Ws + gs(x)@Wn + b), params at d_in[pi]=Ws, [pi+1]=Wn, [pi+2]=b
  auto pp = [&](const float* x, int ei, int N, int pi, int Cin, int Cout,
                float* aggbuf, float* out) {
    gs(x, ei, aggbuf, Cin, N, N);
    lin(x, P(pi), Cin, aggbuf, P(pi + 1), Cin, P(pi + 2), out, N, Cout, 1);
  };

  // ================= stage: inputs -> c1 =================
  {
    off = arena;
    float* xccn = alloc((size_t)BB * NCc * 2);
    float* xfn  = alloc((size_t)BB * NFc * 4);
    inorm(xcc, xccn, NCc, 2);
    inorm(xf,  xfn,  NFc, 4);
    float* aggcf = alloc((size_t)BB * NFc * 2);
    gs(xccn, E_CF, aggcf, 2, NCc, NFc);
    float* h1 = alloc((size_t)BB * NFc * 8);
    lin(aggcf, P(41), 2, nullptr, nullptr, 0, P(42), h1, NFc, 8, 1);     // c1_cf
    float* hcat = alloc((size_t)BB * NFc * 12);
    cat(h1, xfn, hcat, NFc, 8, 4);
    float* aggfp = alloc((size_t)BB * P0 * 12);
    gs(hcat, E_FP, aggfp, 12, NFc, P0);
    float* h2 = alloc((size_t)BB * P0 * 12);
    lin(aggfp, P(43), 12, nullptr, nullptr, 0, P(44), h2, P0, 12, 1);    // c1_fp
    float* agg0 = alloc((size_t)BB * P0 * 12);
    float* ppo  = alloc((size_t)BB * P0 * 12);
    pp(h2, E_PP0, P0, 45, 12, 12, agg0, ppo);                            // c1_pp
    inorm(ppo, c1, P0, 12);
  }
  // ================= stage: c1 -> c2 =================
  {
    off = arena;
    float* pl = alloc((size_t)BB * P1 * 12);
    gs(c1, E_POOL0, pl, 12, P0, P1);
    float* aA = alloc((size_t)BB * P1 * 12); float* hA = alloc((size_t)BB * P1 * 12);
    pp(pl, E_PP1, P1, 48, 12, 12, aA, hA);                               // c2_l1
    float* aB = alloc((size_t)BB * P1 * 12); float* hBv = alloc((size_t)BB * P1 * 12);
    pp(hA, E_PP1, P1, 51, 12, 12, aB, hBv);                              // c2_l2
    inorm(hBv, c2, P1, 12);
  }
  // ================= stage: c2 -> c3 =================
  {
    off = arena;
    float* pl = alloc((size_t)BB * P2 * 12);
    gs(c2, E_POOL1, pl, 12, P1, P2);
    float* aA = alloc((size_t)BB * P2 * 12); float* hA = alloc((size_t)BB * P2 * 12);
    pp(pl, E_PP2, P2, 54, 12, 12, aA, hA);                               // c3_l1
    float* aB = alloc((size_t)BB * P2 * 12); float* hBv = alloc((size_t)BB * P2 * 12);
    pp(hA, E_PP2, P2, 57, 12, 12, aB, hBv);                              // c3_l2
    inorm(hBv, c3, P2, 12);
  }
  // ================= stage: c3 -> c4 (bottleneck + unpool) =================
  {
    off = arena;
    float* pl = alloc((size_t)BB * P3 * 12);
    gs(c3, E_POOL2, pl, 12, P2, P3);
    float* a1 = alloc((size_t)BB * P3 * 12); float* h1 = alloc((size_t)BB * P3 * 24);
    pp(pl, E_PP3, P3, 60, 12, 24, a1, h1);                               // c4_l1
    float* a2 = alloc((size_t)BB * P3 * 24); float* h2 = alloc((size_t)BB * P3 * 24);
    pp(h1, E_PP3, P3, 63, 24, 24, a2, h2);                               // c4_l2
    float* a3 = alloc((size_t)BB * P3 * 24); float* h3 = alloc((size_t)BB * P3 * 24);
    pp(h2, E_PP3, P3, 66, 24, 24, a3, h3);                               // c4_l3
    float* a4 = alloc((size_t)BB * P3 * 24); float* h4 = alloc((size_t)BB * P3 * 24);
    pp(h3, E_PP3, P3, 69, 24, 24, a4, h4);                               // c4_l4
    float* up2 = alloc((size_t)BB * P2 * 24);
    gs(h4, E_UNPOOL2, up2, 24, P3, P2);
    inorm(up2, c4, P2, 24);
  }
  // ================= stage: [c4,c3] -> c7 =================
  {
    off = arena;
    float* ct = alloc((size_t)BB * P2 * 36);
    cat(c4, c3, ct, P2, 24, 12);
    float* aA = alloc((size_t)BB * P2 * 36); float* hA = alloc((size_t)BB * P2 * 24);
    pp(ct, E_PP2, P2, 72, 36, 24, aA, hA);                               // c7_l1
    float* aB = alloc((size_t)BB * P2 * 24); float* hBv = alloc((size_t)BB * P2 * 24);
    pp(hA, E_PP2, P2, 75, 24, 24, aB, hBv);                              // c7_l2
    float* up1 = alloc((size_t)BB * P1 * 24);
    gs(hBv, E_UNPOOL1, up1, 24, P2, P1);
    inorm(up1, c7, P1, 24);
  }
  // ================= stage: [c7,c2] -> c8 =================
  {
    off = arena;
    float* ct = alloc((size_t)BB * P1 * 36);
    cat(c7, c2, ct, P1, 24, 12);
    float* aA = alloc((size_t)BB * P1 * 36); float* hA = alloc((size_t)BB * P1 * 12);
    pp(ct, E_PP1, P1, 78, 36, 12, aA, hA);                               // c8_l1
    float* aB = alloc((size_t)BB * P1 * 12); float* hBv = alloc((size_t)BB * P1 * 12);
    pp(hA, E_PP1, P1, 81, 12, 12, aB, hBv);                              // c8_l2
    float* up0 = alloc((size_t)BB * P0 * 12);
    gs(hBv, E_UNPOOL0, up0, 12, P1, P0);
    inorm(up0, c8, P0, 12);
  }
  // ================= stage: [c8,c1] -> cells -> output =================
  {
    off = arena;
    float* ct = alloc((size_t)BB * P0 * 24);
    cat(c8, c1, ct, P0, 12, 12);
    float* aA = alloc((size_t)BB * P0 * 24); float* hA = alloc((size_t)BB * P0 * 12);
    pp(ct, E_PP0, P0, 84, 24, 12, aA, hA);                               // c9_l1
    float* apc = alloc((size_t)BB * NCc * 12);
    gs(hA, E_PC, apc, 12, P0, NCc);
    float* hcc = alloc((size_t)BB * NCc * 12);
    lin(apc, P(87), 12, nullptr, nullptr, 0, P(88), hcc, NCc, 12, 1);    // c9_l2
    // final: [B,NC,12] @ [12,1] + b  (no relu) -> d_out
    lin(hcc, P(89), 12, nullptr, nullptr, 0, P(90), (float*)d_out, NCc, 1, 0);
  }
  (void)n_in; (void)out_size; (void)ws_size; (void)in_sizes;
}